// DownsamplePoly_70153995813057
// MI455X (gfx1250) — compile-verified
//
#include <hip/hip_runtime.h>

typedef __attribute__((ext_vector_type(2))) float v2f;
typedef __attribute__((ext_vector_type(8))) float v8f;

namespace {
constexpr int T_IN   = 1280000;   // input length (per batch, per channel)
constexpr int N_OUT  = 100000;    // output length
constexpr int CH     = 4;         // channels (contiguous innermost dim)
constexpr int KWIN   = 464;       // x-window per 16-output tile (461 padded to /4)
constexpr int KSTEPS = KWIN / 4;  // 116 wmma 16x16x4 steps
constexpr int TAPS   = 269;       // taps per polyphase branch (1345 = 5*269)
constexpr unsigned XSTRIDE_B8 = 8u * (unsigned)T_IN * CH;  // batch b -> b+8, elements
constexpr unsigned OSTRIDE_B8 = 8u * (unsigned)N_OUT * CH;
}

// s0(n) = ((n+11)*64 - m0)/5 with m0 = ((n+11)*64) mod 5 = (4n+4) mod 5
__device__ __forceinline__ int s_base(int n) {
  int m = (4 * n + 4) % 5;
  return (64 * n + 704 - m) / 5;
}

__device__ __forceinline__ v8f wmma_f32(v2f a, v2f b, v8f c) {
  return __builtin_amdgcn_wmma_f32_16x16x4_f32(false, a, false, b,
                                               (short)0, c, false, false);
}

// x8/out8 are passed as independent kernel args (SGPR bases) so the compiler
// cannot fold them into 64-bit vaddr math; both B streams then share a single
// scaled 32-bit voffset and differ only in the SADDR pair.
__global__ void __launch_bounds__(128)
resample_poly_wmma_f32(const float* __restrict__ x,
                       const float* __restrict__ x8,
                       const float* __restrict__ h,
                       float* __restrict__ out,
                       float* __restrict__ out8)
{
  // 16 x 464 banded-Toeplitz filter block; identical for row tiles 5 apart
  // (s0 shifts by exactly 1024 -> absorbed into xlo).
  __shared__ __align__(16) float Als[16 * KWIN];

  int g     = blockIdx.x;        // 0..3124
  int r     = g % 5;
  int q     = g / 5;             // 0..624
  int tile0 = r + 10 * q;        // row-tile index; tile1 = tile0 + 5

  // ---- build A into LDS ------------------------------------------------
  {
    int n0 = 16 * tile0;
    int sb = s_base(n0);
    for (int idx = threadIdx.x; idx < 16 * KWIN; idx += 128) {
      int i  = idx / KWIN;
      int t  = idx - i * KWIN;
      int ni = n0 + i;
      int mi = (4 * ni + 4) % 5;
      int si = (64 * ni + 704 - mi) / 5;
      int j  = (si - sb) + (TAPS - 1) - t;
      Als[idx] = (j >= 0 && j < TAPS) ? h[mi + 5 * j] : 0.0f;
    }
  }
  __syncthreads();

  int lane   = threadIdx.x & 31;
  int w      = threadIdx.x >> 5;   // 0..3
  int rowSel = w >> 1;             // 0 -> tile0, 1 -> tile0+5
  int cpair  = w & 1;              // accumulators cover batches {4c..4c+3, 4c+8..4c+11}

  int n0  = 16 * (tile0 + 5 * rowSel);
  int xlo = s_base(n0) - (TAPS - 1);

  int col  = lane & 15;            // N index (== A-matrix row M for the LDS read)
  int ksel = (lane >> 4) << 1;     // lane-half -> K offset 0 or 2
  int b0   = cpair * 4 + (col >> 2);
  int ccol = col & 3;

  const float* arow = &Als[col * KWIN];

  v8f acc0 = {0.f,0.f,0.f,0.f,0.f,0.f,0.f,0.f};
  v8f acc1 = {0.f,0.f,0.f,0.f,0.f,0.f,0.f,0.f};

  if (xlo >= 0 && xlo + KWIN <= T_IN) {
    // interior: uniform bases + one 32-bit lane offset; immediates carry 16*kk
    unsigned e0 = (unsigned)(b0 * T_IN + xlo + ksel) * CH + ccol;
#pragma unroll 8
    for (int kk = 0; kk < KSTEPS; ++kk) {
      v2f a   = *(const v2f*)(arow + 4 * kk + ksel);      // ds_load_(2addr_)b64
      v2f bf0 = { x [e0 + 16u * kk], x [e0 + 16u * kk + 4u] };
      v2f bf1 = { x8[e0 + 16u * kk], x8[e0 + 16u * kk + 4u] };
      acc0 = wmma_f32(a, bf0, acc0);
      acc1 = wmma_f32(a, bf1, acc1);
    }
  } else {
    // boundary: branch-free — loads at clamped addresses, zeroed by select
    unsigned rowbase = (unsigned)(b0 * T_IN) * CH + ccol;
#pragma unroll 2
    for (int kk = 0; kk < KSTEPS; ++kk) {
      v2f a  = *(const v2f*)(arow + 4 * kk + ksel);
      int xi = xlo + 4 * kk + ksel;
      int xc0 = min(max(xi, 0), T_IN - 1);
      int xc1 = min(max(xi + 1, 0), T_IN - 1);
      bool ok0 = (unsigned)xi       < (unsigned)T_IN;
      bool ok1 = (unsigned)(xi + 1) < (unsigned)T_IN;
      unsigned i0 = rowbase + (unsigned)xc0 * CH;
      unsigned i1 = rowbase + (unsigned)xc1 * CH;
      float v00 = x [i0];  float v01 = x [i1];
      float v10 = x8[i0];  float v11 = x8[i1];
      v2f bf0 = { ok0 ? v00 : 0.f, ok1 ? v01 : 0.f };
      v2f bf1 = { ok0 ? v10 : 0.f, ok1 ? v11 : 0.f };
      acc0 = wmma_f32(a, bf0, acc0);
      acc1 = wmma_f32(a, bf1, acc1);
    }
  }

  // D layout: VGPR rr -> (M = rr | rr+8 by lane-half, N = lane&15)
  unsigned o0 = (unsigned)(b0 * N_OUT + n0 + ((lane >> 4) << 3)) * CH + ccol;
#pragma unroll
  for (int rr = 0; rr < 8; ++rr) {
    out [o0 + 4u * rr] = acc0[rr];
    out8[o0 + 4u * rr] = acc1[rr];
  }
}

extern "C" void kernel_launch(void* const* d_in, const int* in_sizes, int n_in,
                              void* d_out, int out_size, void* d_ws, size_t ws_size,
                              hipStream_t stream) {
  (void)in_sizes; (void)n_in; (void)d_ws; (void)ws_size; (void)out_size;
  const float* x   = (const float*)d_in[0];
  const float* h   = (const float*)d_in[1];
  float*       out = (float*)d_out;
  const float* x8   = x + XSTRIDE_B8;    // host-side pointer math -> opaque arg
  float*       out8 = out + OSTRIDE_B8;

  // 6250 row tiles of 16 outputs; each block: 2 row tiles x 4 col tiles
  // (4 waves, each wave carries two 16x16 accumulators sharing the A fragment)
  dim3 grid(3125), block(128);
  resample_poly_wmma_f32<<<grid, block, 0, stream>>>(x, x8, h, out, out8);
}